// GAT0_69406671503476
// MI455X (gfx1250) — compile-verified
//
#include <hip/hip_runtime.h>

typedef float v2f __attribute__((ext_vector_type(2)));
typedef float v8f __attribute__((ext_vector_type(8)));

#define N_  4
#define C_  32
#define V_  2048
#define L_  13
#define NC_ (N_ * C_)      // 128
#define J_  (NC_ * L_)     // 1664 columns of the GEMM
#define K_  V_             // 2048 reduction depth
#define BK  32             // K-tile staged in LDS
#define NSTAGES (K_ / BK)  // 64

// ---------------------------------------------------------------------------
// Kernel 1: per-row softmax stats of A2: rowmax and 1/sum(exp(a - rowmax))
// ---------------------------------------------------------------------------
__global__ __launch_bounds__(256)
void gat_rowstats(const float* __restrict__ A,
                  float* __restrict__ smax, float* __restrict__ sinv) {
    __shared__ float red[256];
    const int row = blockIdx.x;
    const int tid = threadIdx.x;
    const float* ar = A + (size_t)row * V_;

    float m = -3.402823466e38f;
    for (int i = tid; i < V_; i += 256) m = fmaxf(m, ar[i]);
    red[tid] = m;
    __syncthreads();
    for (int s = 128; s > 0; s >>= 1) {
        if (tid < s) red[tid] = fmaxf(red[tid], red[tid + s]);
        __syncthreads();
    }
    m = red[0];
    __syncthreads();

    float sum = 0.0f;
    for (int i = tid; i < V_; i += 256) sum += __expf(ar[i] - m);
    red[tid] = sum;
    __syncthreads();
    for (int s = 128; s > 0; s >>= 1) {
        if (tid < s) red[tid] += red[tid + s];
        __syncthreads();
    }
    if (tid == 0) { smax[row] = m; sinv[row] = 1.0f / red[0]; }
}

// ---------------------------------------------------------------------------
// Kernel 2: pack x[n,c,w,l] -> Xp[w, j] with j = (n*C + c)*L + l  (K-major)
// ---------------------------------------------------------------------------
__global__ __launch_bounds__(256)
void gat_pack(const float* __restrict__ x, float* __restrict__ Xp) {
    int gid = blockIdx.x * 256 + threadIdx.x;
    if (gid >= V_ * J_) return;
    int w  = gid / J_;
    int j  = gid - w * J_;
    int nc = j / L_;
    int l  = j - nc * L_;
    Xp[gid] = x[(size_t)nc * (V_ * L_) + w * L_ + l];
}

// ---------------------------------------------------------------------------
// Kernel 3: H = softmax(A2) @ Xp via V_WMMA_F32_16X16X4_F32.
// Workgroup tile 64(M) x 128(N), 4 waves, 32x64 per wave (2x4 WMMA tiles).
// Double-buffered LDS staging; fused epilogue scatter into out[c, n*V+v, l].
// ---------------------------------------------------------------------------
__global__ __launch_bounds__(128)
void gat_gemm(const float* __restrict__ A2, const float* __restrict__ Xp,
              const float* __restrict__ smax, const float* __restrict__ sinv,
              float* __restrict__ out) {
    // A rows stride 34 (even -> aligned b64 frag reads, conflict-free lane
    // stride). B rows stride 144 (2*144 = 32 mod 64 -> the two half-waves of
    // a B-fragment read hit disjoint bank halves).
    __shared__ __align__(16) float ldsA[2][64][34];   // [buf][m][k], softmaxed
    __shared__ __align__(16) float ldsB[2][BK][144];  // [buf][k][j]
    __shared__ float sm[64], si[64];

    const int tid  = threadIdx.x;
    const int wave = tid >> 5;
    const int lane = tid & 31;
    const int half = lane >> 4;   // 0: lanes 0-15, 1: lanes 16-31
    const int l16  = lane & 15;

    const int Mbase = blockIdx.y * 64;
    const int Nbase = blockIdx.x * 128;
    const int waveM = (wave >> 1) * 32;
    const int waveN = (wave & 1) * 64;

    // Cooperative-load coordinates
    const int arow = tid >> 3;          // 0..15 (A tile rows, x4 passes)
    const int acol = (tid & 7) * 4;     // 0..28
    const int brow = tid >> 5;          // 0..3  (B tile rows, x8 passes)
    const int bcol = (tid & 31) * 4;    // 0..124

    if (tid < 64) { sm[tid] = smax[Mbase + tid]; si[tid] = sinv[Mbase + tid]; }
    __syncthreads();

    float4 aReg[4], bReg[8];

    // ---- global -> registers for one K stage ----
    auto loadStage = [&](int k0) {
        #pragma unroll
        for (int p = 0; p < 4; ++p)
            aReg[p] = *(const float4*)(
                &A2[(size_t)(Mbase + arow + p * 16) * V_ + k0 + acol]);
        #pragma unroll
        for (int p = 0; p < 8; ++p)
            bReg[p] = *(const float4*)(
                &Xp[(size_t)(k0 + brow + p * 4) * J_ + Nbase + bcol]);
    };
    // ---- registers -> LDS (A gets softmax applied) ----
    auto storeStage = [&](int buf) {
        #pragma unroll
        for (int p = 0; p < 4; ++p) {
            int r = arow + p * 16;
            float mx = sm[r], iv = si[r];
            ldsA[buf][r][acol + 0] = __expf(aReg[p].x - mx) * iv;
            ldsA[buf][r][acol + 1] = __expf(aReg[p].y - mx) * iv;
            ldsA[buf][r][acol + 2] = __expf(aReg[p].z - mx) * iv;
            ldsA[buf][r][acol + 3] = __expf(aReg[p].w - mx) * iv;
        }
        #pragma unroll
        for (int p = 0; p < 8; ++p) {
            int r = brow + p * 4;
            ldsB[buf][r][bcol + 0] = bReg[p].x;
            ldsB[buf][r][bcol + 1] = bReg[p].y;
            ldsB[buf][r][bcol + 2] = bReg[p].z;
            ldsB[buf][r][bcol + 3] = bReg[p].w;
        }
    };

    v8f acc[2][4] = {};

    // Prologue: stage 0 into buffer 0
    loadStage(0);
    storeStage(0);

    for (int s = 0; s < NSTAGES; ++s) {
        const int buf = s & 1;
        __syncthreads();

        // Kick off next stage's global loads; latency hides under the WMMAs.
        if (s + 1 < NSTAGES) loadStage((s + 1) * BK);

        const float (*lA)[34]  = ldsA[buf];
        const float (*lB)[144] = ldsB[buf];

        #pragma unroll
        for (int kk = 0; kk < BK; kk += 4) {
            const int fk = kk + half * 2;   // K pair owned by this half-wave
            v2f aF[2], bF[4];
            #pragma unroll
            for (int mt = 0; mt < 2; ++mt)
                aF[mt] = *(const v2f*)(&lA[waveM + mt * 16 + l16][fk]);
            #pragma unroll
            for (int nt = 0; nt < 4; ++nt) {
                int bc = waveN + nt * 16 + l16;
                v2f b;
                b.x = lB[fk + 0][bc];
                b.y = lB[fk + 1][bc];
                bF[nt] = b;
            }
            #pragma unroll
            for (int mt = 0; mt < 2; ++mt)
                #pragma unroll
                for (int nt = 0; nt < 4; ++nt)
                    acc[mt][nt] = __builtin_amdgcn_wmma_f32_16x16x4_f32(
                        false, aF[mt], false, bF[nt],
                        (short)0, acc[mt][nt], false, false);
        }

        if (s + 1 < NSTAGES) storeStage(buf ^ 1);
    }

    // ---- epilogue: H[v, j] -> out[c*(N*V*L) + n*(V*L) + v*L + l] ----
    #pragma unroll
    for (int nt = 0; nt < 4; ++nt) {
        int j  = Nbase + waveN + nt * 16 + l16;
        int nc = j / L_;
        int l  = j - nc * L_;
        int n  = nc >> 5;      // nc / C_
        int c  = nc & 31;      // nc % C_
        float* op = out + (size_t)c * (N_ * V_ * L_) + (size_t)n * (V_ * L_) + l;
        #pragma unroll
        for (int mt = 0; mt < 2; ++mt) {
            int vbase = Mbase + waveM + mt * 16 + half * 8;  // C/D: M = r + 8*half
            #pragma unroll
            for (int r = 0; r < 8; ++r)
                op[(size_t)(vbase + r) * L_] = acc[mt][nt][r];
        }
    }
}

// ---------------------------------------------------------------------------
extern "C" void kernel_launch(void* const* d_in, const int* in_sizes, int n_in,
                              void* d_out, int out_size, void* d_ws, size_t ws_size,
                              hipStream_t stream) {
    const float* x  = (const float*)d_in[0];
    const float* A1 = (const float*)d_in[1];   // dead in the reference
    const float* A2 = (const float*)d_in[2];
    float* out = (float*)d_out;
    (void)A1; (void)in_sizes; (void)n_in; (void)out_size; (void)ws_size;

    float* smax = (float*)d_ws;                          // 2048 floats
    float* sinv = smax + V_;                             // 2048 floats
    float* Xp   = (float*)((char*)d_ws + 16384);         // 2048*1664 floats

    gat_rowstats<<<V_, 256, 0, stream>>>(A2, smax, sinv);

    gat_pack<<<(V_ * J_ + 255) / 256, 256, 0, stream>>>(x, Xp);

    dim3 grid(J_ / 128, V_ / 64);   // 13 x 32 workgroups
    gat_gemm<<<grid, 128, 0, stream>>>(A2, Xp, smax, sinv, out);
}